// VectorQuantizer_30640296690494
// MI455X (gfx1250) — compile-verified
//
#include <hip/hip_runtime.h>
#include <hip/hip_bf16.h>

// ---------------------------------------------------------------------------
// Problem constants (from reference)
// ---------------------------------------------------------------------------
#define BATCH    32
#define EDIM     256
#define LATW     16
#define NPOS     256              // LATW*LATW
#define NROWS    8192             // BATCH * NPOS
#define NE       16384            // codebook entries
#define KSEL     256              // top-k

// Output layout (concatenated flat, reference return order)
static constexpr size_t OFF_ZQ   = 0;                          // 2,097,152
static constexpr size_t OFF_PERP = 2097152ull;                 // 1
static constexpr size_t OFF_ENC  = 2097153ull;                 // 134,217,728
static constexpr size_t OFF_IDX  = 136314881ull;               // 2,097,152
static constexpr size_t OFF_D    = 138412033ull;               // 134,217,728
static constexpr size_t OFF_EMB  = 272629761ull;               // 4,194,304

// Workspace layout (floats/uints)
static constexpr size_t WS_ZN     = 0;                         // 8192 f32
static constexpr size_t WS_EN     = 8192;                      // 16384 f32
static constexpr size_t WS_COUNTS = 8192 + 16384;              // 16384 u32

typedef float v2f __attribute__((ext_vector_type(2)));
typedef float v8f __attribute__((ext_vector_type(8)));

// ---------------------------------------------------------------------------
// 1) z row norms: zn[n] = sum_c z[b,c,h,w]^2 , n = b*256 + (h*16+w)
// ---------------------------------------------------------------------------
__global__ __launch_bounds__(256) void znorm_kernel(const float* __restrict__ z,
                                                    float* __restrict__ zn) {
  int n = blockIdx.x * 256 + threadIdx.x;        // 0..8191
  int b = n >> 8, p = n & 255;
  const float* base = z + (size_t)b * (EDIM * NPOS) + p;
  float s = 0.f;
#pragma unroll 8
  for (int c = 0; c < EDIM; ++c) {
    float v = base[(size_t)c * NPOS];
    s += v * v;
  }
  zn[n] = s;
}

// ---------------------------------------------------------------------------
// 2) e row norms (wave per row) fused with emb_weight pass-through copy
// ---------------------------------------------------------------------------
__global__ __launch_bounds__(256) void enorm_kernel(const float* __restrict__ emb,
                                                    float* __restrict__ en,
                                                    float* __restrict__ embcopy) {
  int wave = threadIdx.x >> 5, lane = threadIdx.x & 31;
  int row = blockIdx.x * 8 + wave;               // 0..16383
  const float* r = emb + (size_t)row * EDIM;
  float s = 0.f;
#pragma unroll
  for (int i = lane; i < EDIM; i += 32) {
    float v = r[i];
    s += v * v;
    embcopy[(size_t)row * EDIM + i] = v;
  }
#pragma unroll
  for (int o = 16; o > 0; o >>= 1) s += __shfl_down(s, o, 32);
  if (lane == 0) en[row] = s;
}

// ---------------------------------------------------------------------------
// 3) zero the one-hot region (scalar: base offset is only 4B aligned)
// ---------------------------------------------------------------------------
__global__ __launch_bounds__(256) void zero_enc_kernel(float* __restrict__ enc, size_t count) {
  size_t i = (size_t)blockIdx.x * 256 + threadIdx.x;
  size_t stride = (size_t)gridDim.x * 256;
  for (; i < count; i += stride) enc[i] = 0.0f;
}

__global__ __launch_bounds__(256) void zero_counts_kernel(unsigned* __restrict__ counts) {
  int i = blockIdx.x * 256 + threadIdx.x;
  if (i < NE) counts[i] = 0u;
}

// ---------------------------------------------------------------------------
// 4) d = ||z||^2 + ||e||^2 - 2 z.e^T  via fp32 WMMA (16x16x4)
//    128x128 tile / block (8 wave32), BK=32.
//    A tile staged with global_load_async_to_lds_b128 (raw k-major image);
//    B tile staged through VGPRs, pre-paired into the WMMA fragment layout.
// ---------------------------------------------------------------------------
__global__ __launch_bounds__(256) void gemm_d_kernel(const float* __restrict__ z,
                                                     const float* __restrict__ emb,
                                                     const float* __restrict__ zn,
                                                     const float* __restrict__ en,
                                                     float* __restrict__ dout) {
  __shared__ float Alds[32][128];      // [k][m] raw image of the A tile
  __shared__ v2f   Blds2[16][130];     // [k/2][n] pre-paired (k,k+1), padded

  const int tm  = blockIdx.y;          // 0..63  (M tiles of 128 rows)
  const int tn  = blockIdx.x;          // 0..127 (N tiles of 128 codes)
  const int tid = threadIdx.x;
  const int wave = tid >> 5;
  const int lane = tid & 31;
  const int l16  = lane & 15;
  const int hi   = lane >> 4;          // 0 | 1

  const int row0 = tm * 128;           // flat latent row base (inside one batch)
  const int b    = row0 >> 8;
  const int p0   = row0 & 255;
  const int col0 = tn * 128;

  v8f acc[8];
#pragma unroll
  for (int j = 0; j < 8; ++j) acc[j] = (v8f){0.f,0.f,0.f,0.f,0.f,0.f,0.f,0.f};

  const int a_kk  = tid >> 3;          // 0..31
  const int a_r   = (tid & 7) * 16;    // 0..112
  const int b_col = tid >> 1;          // 0..127
  const int b_kh  = (tid & 1) * 16;    // 0 | 16

  const unsigned ldsA = (unsigned)(uintptr_t)&Alds[a_kk][a_r];

  for (int k0 = 0; k0 < EDIM; k0 += 32) {
    // --- A tile: async copy, memory image == LDS image (k-major, contig n) ---
    {
      const float* srcA = z + (size_t)b * (EDIM * NPOS) + (size_t)(k0 + a_kk) * NPOS + p0 + a_r;
      asm volatile(
          "global_load_async_to_lds_b128 %0, %1, off\n\t"
          "global_load_async_to_lds_b128 %0, %1, off offset:16\n\t"
          "global_load_async_to_lds_b128 %0, %1, off offset:32\n\t"
          "global_load_async_to_lds_b128 %0, %1, off offset:48"
          :: "v"(ldsA), "v"(srcA) : "memory");
    }
    // --- B tile: B[k][n] = emb[col][k]; pair (k,k+1) per v2f for ds_load_b64 feed ---
    {
      const float* src = emb + (size_t)(col0 + b_col) * EDIM + k0 + b_kh;
      float4 b0 = *(const float4*)(src + 0);
      float4 b1 = *(const float4*)(src + 4);
      float4 b2 = *(const float4*)(src + 8);
      float4 b3 = *(const float4*)(src + 12);
      const int kb = b_kh >> 1;        // 0 or 8
      Blds2[kb + 0][b_col] = (v2f){b0.x, b0.y};
      Blds2[kb + 1][b_col] = (v2f){b0.z, b0.w};
      Blds2[kb + 2][b_col] = (v2f){b1.x, b1.y};
      Blds2[kb + 3][b_col] = (v2f){b1.z, b1.w};
      Blds2[kb + 4][b_col] = (v2f){b2.x, b2.y};
      Blds2[kb + 5][b_col] = (v2f){b2.z, b2.w};
      Blds2[kb + 6][b_col] = (v2f){b3.x, b3.y};
      Blds2[kb + 7][b_col] = (v2f){b3.z, b3.w};
    }
    // prefetch next k-stage of B (global_prefetch_b8)
    if (k0 + 32 < EDIM)
      __builtin_prefetch(emb + (size_t)(col0 + b_col) * EDIM + k0 + 32 + b_kh, 0, 1);

    // async A copies tracked with ASYNCcnt (not covered by the barrier waits)
    asm volatile("s_wait_asynccnt 0x0" ::: "memory");
    __syncthreads();

    const int arow = wave * 16 + l16;
#pragma unroll
    for (int kk2 = 0; kk2 < 16; kk2 += 2) {     // k advances 4 per step
      v2f a;
      a.x = Alds[2 * kk2 + 2 * hi + 0][arow];
      a.y = Alds[2 * kk2 + 2 * hi + 1][arow];
#pragma unroll
      for (int j = 0; j < 8; ++j) {
        v2f bb = Blds2[kk2 + hi][j * 16 + l16];  // single ds_load_b64
        acc[j] = __builtin_amdgcn_wmma_f32_16x16x4_f32(
            /*neg_a=*/false, a, /*neg_b=*/false, bb,
            /*c_mod=*/(short)0, acc[j], /*reuse_a=*/false, /*reuse_b=*/false);
      }
    }
    __syncthreads();
  }

  // epilogue: d = zn + en - 2*acc ; C/D layout: VGPR r -> M = r + 8*hi, lane -> N
#pragma unroll
  for (int j = 0; j < 8; ++j) {
    const int cg = col0 + j * 16 + l16;
    const float e = en[cg];
#pragma unroll
    for (int r = 0; r < 8; ++r) {
      const int rg = row0 + wave * 16 + r + 8 * hi;
      dout[(size_t)rg * NE + cg] = zn[rg] + e - 2.0f * acc[j][r];
    }
  }
}

// ---------------------------------------------------------------------------
// 5) per-row top-256 (ascending) via radix-select + bitonic sort,
//    fused scatter of one-hots, index output, usage histogram, z_q (+pos)
// ---------------------------------------------------------------------------
__global__ __launch_bounds__(256) void topk_kernel(const float* __restrict__ dmat,
                                                   const float* __restrict__ emb,
                                                   const float* __restrict__ pos,
                                                   float* __restrict__ out,
                                                   unsigned* __restrict__ counts) {
  __shared__ unsigned hist[256];
  __shared__ unsigned long long cand[512];
  __shared__ unsigned scnt;
  __shared__ unsigned sidx[256];
  __shared__ unsigned sprefix, skrem;

  const int row = blockIdx.x;            // 0..8191
  const int t   = threadIdx.x;           // 0..255
  const float* drow = dmat + (size_t)row * NE;

  // 64 strided elements per thread, kept in registers; order-preserving key
  unsigned key[64];
#pragma unroll
  for (int i = 0; i < 64; ++i) {
    unsigned u = __float_as_uint(drow[i * 256 + t]);
    key[i] = (u & 0x80000000u) ? ~u : (u | 0x80000000u);
  }

  // ---- radix select: value of the 256th smallest key ----
  unsigned prefix = 0, krem = KSEL;
  for (int pass = 0; pass < 4; ++pass) {
    const int shift = 24 - 8 * pass;
    const unsigned hmask = (pass == 0) ? 0u : (0xFFFFFFFFu << (shift + 8));
    hist[t] = 0u;
    __syncthreads();
#pragma unroll
    for (int i = 0; i < 64; ++i)
      if ((key[i] & hmask) == (prefix & hmask))
        atomicAdd(&hist[(key[i] >> shift) & 255u], 1u);
    __syncthreads();
    if (t == 0) {
      unsigned cum = 0, bsel = 0;
      for (; bsel < 256; ++bsel) {
        unsigned h = hist[bsel];
        if (cum + h >= krem) break;
        cum += h;
      }
      sprefix = prefix | (bsel << shift);
      skrem   = krem - cum;
    }
    __syncthreads();
    prefix = sprefix;
    krem   = skrem;
    __syncthreads();
  }
  const unsigned T = prefix;             // exact key of the k-th smallest

  // ---- compact candidates (key <= T) into LDS as (key<<32)|index ----
  if (t == 0) scnt = 0u;
  __syncthreads();
#pragma unroll
  for (int i = 0; i < 64; ++i) {
    if (key[i] <= T) {
      unsigned slot = atomicAdd(&scnt, 1u);
      if (slot < 512u)
        cand[slot] = ((unsigned long long)key[i] << 32) | (unsigned)(i * 256 + t);
    }
  }
  __syncthreads();
  const unsigned total = scnt < 512u ? scnt : 512u;
  for (int s = t; s < 512; s += 256)
    if ((unsigned)s >= total) cand[s] = 0xFFFFFFFFFFFFFFFFull;
  __syncthreads();

  // ---- bitonic sort of 512 u64 keys ascending (ties break by index) ----
  for (int size = 2; size <= 512; size <<= 1) {
    for (int stride = size >> 1; stride > 0; stride >>= 1) {
      int i = 2 * t - (t & (stride - 1));
      bool asc = ((i & size) == 0);
      unsigned long long x = cand[i], y = cand[i + stride];
      if ((x > y) == asc) { cand[i] = y; cand[i + stride] = x; }
      __syncthreads();
    }
  }

  // ---- outputs: indices (as float), one-hot scatter, usage histogram ----
  const unsigned idx = (unsigned)(cand[t] & 0xFFFFFFFFull);
  sidx[t] = idx;
  out[OFF_IDX + (size_t)row * KSEL + t] = (float)idx;
  out[OFF_ENC + (size_t)row * NE + idx] = 1.0f;
  atomicAdd(&counts[idx], 1u);
  __syncthreads();

  // ---- z_q[row][c] = sum_j emb[sidx[j]][c] + pos[row%256][c], transposed store ----
  float s = 0.f;
  for (int j = 0; j < KSEL; ++j) s += emb[(size_t)sidx[j] * EDIM + t];
  const int p = row & 255, bb = row >> 8;
  s += pos[p * EDIM + t];
  out[OFF_ZQ + (size_t)bb * (EDIM * NPOS) + (size_t)t * NPOS + p] = s;
}

// ---------------------------------------------------------------------------
// 6) perplexity from global usage counts
// ---------------------------------------------------------------------------
__global__ __launch_bounds__(256) void perplexity_kernel(const unsigned* __restrict__ counts,
                                                         float* __restrict__ out) {
  __shared__ float red[256];
  const int t = threadIdx.x;
  float s = 0.f;
  for (int i = t; i < NE; i += 256) {
    float e = (float)counts[i] * (1.0f / (float)NROWS);
    s += e * __logf(e + 1e-10f);
  }
  red[t] = s;
  __syncthreads();
  for (int o = 128; o > 0; o >>= 1) {
    if (t < o) red[t] += red[t + o];
    __syncthreads();
  }
  if (t == 0) out[OFF_PERP] = __expf(-red[0]);
}

// ---------------------------------------------------------------------------
// launch
// ---------------------------------------------------------------------------
extern "C" void kernel_launch(void* const* d_in, const int* in_sizes, int n_in,
                              void* d_out, int out_size, void* d_ws, size_t ws_size,
                              hipStream_t stream) {
  const float* z    = (const float*)d_in[0];   // (32,256,16,16)
  const float* emb  = (const float*)d_in[1];   // (16384,256)
  const float* pos  = (const float*)d_in[2];   // (256,256)
  float* out = (float*)d_out;

  float*    ws_f    = (float*)d_ws;
  float*    zn      = ws_f + WS_ZN;
  float*    en      = ws_f + WS_EN;
  unsigned* counts  = (unsigned*)(ws_f + WS_COUNTS);

  znorm_kernel<<<NROWS / 256, 256, 0, stream>>>(z, zn);
  enorm_kernel<<<NE / 8, 256, 0, stream>>>(emb, en, out + OFF_EMB);
  zero_enc_kernel<<<4096, 256, 0, stream>>>(out + OFF_ENC, (size_t)NROWS * NE);
  zero_counts_kernel<<<NE / 256, 256, 0, stream>>>(counts);

  gemm_d_kernel<<<dim3(NE / 128, NROWS / 128), 256, 0, stream>>>(z, emb, zn, en, out + OFF_D);

  topk_kernel<<<NROWS, 256, 0, stream>>>(out + OFF_D, emb, pos, out, counts);
  perplexity_kernel<<<1, 256, 0, stream>>>(counts, out);
}